// InteractionPropertyHead_24266565222501
// MI455X (gfx1250) — compile-verified
//
#include <hip/hip_runtime.h>
#include <math.h>
#include <stdint.h>

typedef float v2f __attribute__((ext_vector_type(2)));
typedef float v8f __attribute__((ext_vector_type(8)));
typedef unsigned int u32x4 __attribute__((ext_vector_type(4)));
typedef int i32x8 __attribute__((ext_vector_type(8)));
typedef int i32x4 __attribute__((ext_vector_type(4)));

#define NVAR 512
#define HDIM 512
#define PDIM 8

#if __has_builtin(__builtin_amdgcn_tensor_load_to_lds)
#define USE_TDM 1
#else
#define USE_TDM 0
#endif

// ---------------------------------------------------------------------------
// GEMM (NT): C[M x Ncols] = scale * (A[M x K] @ W[Ncols x K]^T) + bias[col]
// One wave32 computes a 16x32 strip (two 16x16 tiles) via
// V_WMMA_F32_16X16X4_F32, reusing the A fragment across both column tiles
// (3 float2 loads -> 2 WMMAs per K-step instead of 2 loads -> 1 WMMA).
// A layout (ISA 7.12.2): lanes 0-15 hold M=lane with K=k+0/k+1 in 2 VGPRs,
// lanes 16-31 hold K=k+2/k+3 -> one aligned float2 load per operand per step.
// C/D: VGPR r holds row r + 8*half, column = lane&15 of the tile.
// ---------------------------------------------------------------------------
__global__ void __launch_bounds__(256)
gemm_nt_wmma_f32(const float* __restrict__ A, int lda,
                 const float* __restrict__ W, int ldw, int wcoloff,
                 const float* __restrict__ bias,
                 float* __restrict__ C, int ldc,
                 int ntileColPairs, int K, float scale)
{
    const int lane = threadIdx.x & 31;
    const int wid  = threadIdx.x >> 5;
    const int task = blockIdx.x * 8 + wid;
    const int trow = task / ntileColPairs;
    const int tcp  = task - trow * ntileColPairs;
    const int half = lane >> 4;          // 0: K=k+0/1,  1: K=k+2/3
    const int lm   = lane & 15;

    const int arow  = trow * 16 + lm;    // M index held by this lane (A)
    const int bcol0 = tcp * 32 + lm;     // N index, first column tile
    const int bcol1 = bcol0 + 16;        // N index, second column tile

    const float* aptr  = A + (size_t)arow  * lda + 2 * half;
    const float* bptr0 = W + (size_t)bcol0 * ldw + wcoloff + 2 * half;
    const float* bptr1 = W + (size_t)bcol1 * ldw + wcoloff + 2 * half;

    v8f c0 = {}, c1 = {};
    for (int k = 0; k < K; k += 4) {
        v2f av  = *(const v2f*)(aptr  + k);
        v2f bv0 = *(const v2f*)(bptr0 + k);
        v2f bv1 = *(const v2f*)(bptr1 + k);
        c0 = __builtin_amdgcn_wmma_f32_16x16x4_f32(false, av, false, bv0,
                                                   (short)0, c0, false, false);
        c1 = __builtin_amdgcn_wmma_f32_16x16x4_f32(false, av, false, bv1,
                                                   (short)0, c1, false, false);
    }

    const float bval0 = bias ? bias[bcol0] : 0.0f;
    const float bval1 = bias ? bias[bcol1] : 0.0f;
    #pragma unroll
    for (int r = 0; r < 8; ++r) {
        int row = trow * 16 + r + 8 * half;
        C[(size_t)row * ldc + bcol0] = c0[r] * scale + bval0;
        C[(size_t)row * ldc + bcol1] = c1[r] * scale + bval1;
    }
}

// ---------------------------------------------------------------------------
// Masked row softmax: one block (256 threads) per row of the 512-wide matrix.
// ---------------------------------------------------------------------------
__global__ void __launch_bounds__(256)
softmax_mask_kernel(const float* __restrict__ attn,
                    const unsigned char* __restrict__ mask,
                    float* __restrict__ out, int N)
{
    __shared__ float red[256];
    const int i   = blockIdx.x;
    const int tid = threadIdx.x;
    const bool rowvalid = mask[i] != 0;

    float a0 = attn[(size_t)i * N + tid];
    float a1 = attn[(size_t)i * N + tid + 256];
    if (!(rowvalid && mask[tid]       != 0)) a0 = -INFINITY;
    if (!(rowvalid && mask[tid + 256] != 0)) a1 = -INFINITY;

    red[tid] = fmaxf(a0, a1);
    __syncthreads();
    #pragma unroll
    for (int s = 128; s > 0; s >>= 1) {
        if (tid < s) red[tid] = fmaxf(red[tid], red[tid + s]);
        __syncthreads();
    }
    const float m = red[0];
    __syncthreads();

    float e0 = (a0 == -INFINITY) ? 0.0f : expf(a0 - m);
    float e1 = (a1 == -INFINITY) ? 0.0f : expf(a1 - m);
    red[tid] = e0 + e1;
    __syncthreads();
    #pragma unroll
    for (int s = 128; s > 0; s >>= 1) {
        if (tid < s) red[tid] += red[tid + s];
        __syncthreads();
    }
    const float inv = 1.0f / red[0];

    out[(size_t)i * N + tid]       = e0 * inv;
    out[(size_t)i * N + tid + 256] = e1 * inv;
}

// ---------------------------------------------------------------------------
// Pair head: one wave32 per (i,j) pair, 8 pairs per block.
// h = A[i] + B[j] (b1 pre-folded into A), LayerNorm, exact-erf GELU,
// logits = h . W2[p] + b2, sigmoid. Wave-level butterfly reductions (wave32).
// W2 (8x512 fp32, 16 KB) is staged into LDS by the Tensor Data Mover:
// one TENSOR_LOAD_TO_LDS issued by wave 0 (scalar branch; TDM ignores EXEC),
// drained with s_wait_tensorcnt 0 before the workgroup barrier.
// ---------------------------------------------------------------------------
__device__ __forceinline__ float wave_sum(float v) {
    #pragma unroll
    for (int off = 16; off > 0; off >>= 1) v += __shfl_xor(v, off, 32);
    return v;
}

__global__ void __launch_bounds__(256)
pair_head_kernel(const float* __restrict__ Arow, const float* __restrict__ Brow,
                 const float* __restrict__ ln_g, const float* __restrict__ ln_b,
                 const float* __restrict__ W2,   const float* __restrict__ b2,
                 float* __restrict__ logits, float* __restrict__ probs, int N)
{
    __shared__ float sW2[PDIM * HDIM];   // 16 KB, filled by TDM
    __shared__ float sG[HDIM];
    __shared__ float sB[HDIM];
    __shared__ float sB2[PDIM];

    const int tid = threadIdx.x;

#if USE_TDM
    // Wave 0 issues the tensor DMA for the whole block (scalar-uniform branch).
    if (__builtin_amdgcn_readfirstlane(tid >> 5) == 0) {
        unsigned long long ga = (unsigned long long)(uintptr_t)W2;
        unsigned int la = (unsigned int)(uintptr_t)&sW2[0]; // LDS byte offset = addr[31:0]
        u32x4 g0;
        g0[0] = 1u;                                         // count=1 (valid user D#)
        g0[1] = la;                                         // lds_addr
        g0[2] = (unsigned int)ga;                           // global_addr[31:0]
        g0[3] = ((unsigned int)(ga >> 32) & 0x01FFFFFFu)    // global_addr[56:32]
              | (2u << 30);                                 // type=2 ("image")
        i32x8 g1;
        g1[0] = (int)(2u << 16);     // workgroup_mask=0, data_size=2 (4 bytes)
        g1[1] = (int)(512u << 16);   // tensor_dim0 = 512 (bits [79:48], low half)
        g1[2] = (int)(8u << 16);     // tensor_dim0 hi = 0, tensor_dim1 = 8
        g1[3] = (int)(512u << 16);   // tensor_dim1 hi = 0, tile_dim0 = 512
        g1[4] = 8;                   // tile_dim1 = 8, tile_dim2 = 0 (2D tile)
        g1[5] = 512;                 // tensor_dim0_stride = 512 elements
        g1[6] = 0;                   // stride hi / dim1_stride lo
        g1[7] = 0;                   // dim1_stride hi
        i32x4 gz4 = {0, 0, 0, 0};            // groups 2/3 unused for a 2D tensor
        i32x8 gz8 = {0, 0, 0, 0, 0, 0, 0, 0};
        __builtin_amdgcn_tensor_load_to_lds(g0, g1, gz4, gz4, gz8, 0);
        __builtin_amdgcn_s_wait_tensorcnt(0);
    }
#else
    for (int idx = tid; idx < PDIM * HDIM; idx += 256) sW2[idx] = W2[idx];
#endif
    for (int idx = tid; idx < HDIM; idx += 256) { sG[idx] = ln_g[idx]; sB[idx] = ln_b[idx]; }
    if (tid < PDIM) sB2[tid] = b2[tid];
    __syncthreads();

    const int lane = tid & 31;
    const int wid  = tid >> 5;
    const int pair = blockIdx.x * 8 + wid;
    const int i    = pair / N;
    const int j    = pair - i * N;

    const float* ai = Arow + (size_t)i * HDIM;
    const float* bj = Brow + (size_t)j * HDIM;

    float h[HDIM / 32];
    float s = 0.0f, sq = 0.0f;
    #pragma unroll
    for (int k = 0; k < HDIM / 32; ++k) {
        float v = ai[lane + 32 * k] + bj[lane + 32 * k];
        h[k] = v;
        s  += v;
        sq += v * v;
    }
    s  = wave_sum(s);
    sq = wave_sum(sq);
    const float mu  = s * (1.0f / HDIM);
    const float var = sq * (1.0f / HDIM) - mu * mu;
    const float rs  = rsqrtf(var + 1e-5f);

    float acc[PDIM];
    #pragma unroll
    for (int p = 0; p < PDIM; ++p) acc[p] = 0.0f;

    #pragma unroll
    for (int k = 0; k < HDIM / 32; ++k) {
        const int idx = lane + 32 * k;
        float x = (h[k] - mu) * rs * sG[idx] + sB[idx];
        float g = 0.5f * x * (1.0f + erff(x * 0.70710678118654752f));  // exact GELU
        #pragma unroll
        for (int p = 0; p < PDIM; ++p) acc[p] += g * sW2[p * HDIM + idx];
    }

    float myout = 0.0f;
    #pragma unroll
    for (int p = 0; p < PDIM; ++p) {
        float r = wave_sum(acc[p]);
        if (lane == p) myout = r;
    }
    if (lane < PDIM) {
        const size_t base = ((size_t)i * N + j) * PDIM + lane;
        float v = myout + sB2[lane];
        logits[base] = v;
        probs[base]  = 1.0f / (1.0f + expf(-v));
    }
}

// ---------------------------------------------------------------------------
extern "C" void kernel_launch(void* const* d_in, const int* in_sizes, int n_in,
                              void* d_out, int out_size, void* d_ws, size_t ws_size,
                              hipStream_t stream) {
    const float*         E    = (const float*)d_in[0];
    const unsigned char* mask = (const unsigned char*)d_in[1];
    const float*         Wq   = (const float*)d_in[2];
    const float*         bq   = (const float*)d_in[3];
    const float*         Wk   = (const float*)d_in[4];
    const float*         bk   = (const float*)d_in[5];
    const float*         W1   = (const float*)d_in[6];
    const float*         b1   = (const float*)d_in[7];
    const float*         ln_g = (const float*)d_in[8];
    const float*         ln_b = (const float*)d_in[9];
    const float*         W2   = (const float*)d_in[10];
    const float*         b2   = (const float*)d_in[11];

    const int N = NVAR, H = HDIM, P = PDIM;

    float* ws    = (float*)d_ws;
    float* Qm    = ws;                 // [N,H]
    float* Km    = ws + 1 * N * H;     // [N,H]
    float* Am    = ws + 2 * N * H;     // [N,H]  (b1 folded in)
    float* Bm    = ws + 3 * N * H;     // [N,H]
    float* attn  = ws + 4 * N * H;     // [N,N] pre-softmax

    float* out    = (float*)d_out;
    float* logits = out;                         // [N,N,P]
    float* probs  = out + (size_t)N * N * P;     // [N,N,P]
    float* attnw  = out + (size_t)2 * N * N * P; // [N,N]

    const int ntcp  = N / 32;                    // 16 column-pair strips
    const int tasks = (N / 16) * ntcp;           // 512 wave-tasks
    dim3 gblk(tasks / 8), tblk(256);

    // Q = E @ Wq^T + bq ; K = E @ Wk^T + bk
    gemm_nt_wmma_f32<<<gblk, tblk, 0, stream>>>(E, H, Wq, H, 0, bq, Qm, H, ntcp, H, 1.0f);
    gemm_nt_wmma_f32<<<gblk, tblk, 0, stream>>>(E, H, Wk, H, 0, bk, Km, H, ntcp, H, 1.0f);
    // A = E @ W1[:, :H]^T + b1 (fold b1) ; B = E @ W1[:, H:]^T
    gemm_nt_wmma_f32<<<gblk, tblk, 0, stream>>>(E, H, W1, 2 * H, 0, b1, Am, H, ntcp, H, 1.0f);
    gemm_nt_wmma_f32<<<gblk, tblk, 0, stream>>>(E, H, W1, 2 * H, H, nullptr, Bm, H, ntcp, H, 1.0f);
    // attn = (Q @ K^T) / sqrt(H)
    gemm_nt_wmma_f32<<<gblk, tblk, 0, stream>>>(Qm, H, Km, H, 0, nullptr, attn, N, ntcp, H,
                                                1.0f / sqrtf((float)H));
    // softmax with mask -> attn_weights output
    softmax_mask_kernel<<<N, 256, 0, stream>>>(attn, mask, attnw, N);
    // pairwise MLP head: LN + GELU + W2 projection + sigmoid
    pair_head_kernel<<<(N * N) / 8, 256, 0, stream>>>(Am, Bm, ln_g, ln_b, W2, b2,
                                                      logits, probs, N);
}